// HyperbolicExponential_13048110645422
// MI455X (gfx1250) — compile-verified
//
#include <hip/hip_runtime.h>

// Hyperbolic exponential map, N x 32 f32 rows.
// MI455X strategy: pure HBM stream (768 MB total, single-touch) ->
//   - one full 128B row per lane: all Minkowski reductions are local fma
//     chains, zero cross-lane traffic, minimal wave-instruction count
//   - aligned global_load_b128 / global_store_b128 with NON-TEMPORAL hints
//     (working set is 4x the 192MB L2; nothing is reused)
//   - the first hyperboloid projection's scale factor is EXACTLY 0 or 1,
//     so it is folded into scalar coefficients (bit-exact) instead of a
//     32-wide vector multiply -> ~23 fewer VALU per wave
//   - wave32-native: 256-thread blocks = 8 waves, 256 rows per block

typedef __attribute__((ext_vector_type(4))) float v4f;

#define HEPS   1e-8f
#define HMAXN  20.0f

// r = a*p + q   (componentwise fma)
__device__ __forceinline__ v4f v4_fma(float a, const v4f p, const v4f q) {
  v4f r;
  r.x = fmaf(a, p.x, q.x);
  r.y = fmaf(a, p.y, q.y);
  r.z = fmaf(a, p.z, q.z);
  r.w = fmaf(a, p.w, q.w);
  return r;
}

// r = a*p + b*q
__device__ __forceinline__ v4f v4_ab(float a, const v4f p, float b, const v4f q) {
  v4f r;
  r.x = fmaf(a, p.x, b * q.x);
  r.y = fmaf(a, p.y, b * q.y);
  r.z = fmaf(a, p.z, b * q.z);
  r.w = fmaf(a, p.w, b * q.w);
  return r;
}

__device__ __forceinline__ v4f v4_mul(const v4f p, float a) {
  v4f r;
  r.x = p.x * a; r.y = p.y * a; r.z = p.z * a; r.w = p.w * a;
  return r;
}

// sum over the 31 "space" components of A[i]*B[i] (skips slot 0.x = time)
__device__ __forceinline__ float spacedot(const v4f* A, const v4f* B) {
  float s = A[0].y * B[0].y;
  s = fmaf(A[0].z, B[0].z, s);
  s = fmaf(A[0].w, B[0].w, s);
#pragma unroll
  for (int i = 1; i < 8; ++i) {
    s = fmaf(A[i].x, B[i].x, s);
    s = fmaf(A[i].y, B[i].y, s);
    s = fmaf(A[i].z, B[i].z, s);
    s = fmaf(A[i].w, B[i].w, s);
  }
  return s;
}

// full Minkowski inner product on a materialized vector: -a0*b0 + space dot
__device__ __forceinline__ float mink(const v4f* A, const v4f* B) {
  float s = spacedot(A, B);
  s = fmaf(-A[0].x, B[0].x, s);
  return s;
}

__global__ __launch_bounds__(256) void hyperbolic_expmap_kernel(
    const float* __restrict__ xg, const float* __restrict__ vg,
    float* __restrict__ og, long long nrows) {
  const long long row = (long long)blockIdx.x * blockDim.x + threadIdx.x;
  if (row >= nrows) return;

  const v4f* xp = (const v4f*)xg + row * 8;   // 128B-aligned row
  const v4f* vp = (const v4f*)vg + row * 8;
  v4f*       op = (v4f*)og + row * 8;

  v4f X[8], V[8];
#pragma unroll
  for (int i = 0; i < 8; ++i) X[i] = __builtin_nontemporal_load(xp + i);
#pragma unroll
  for (int i = 0; i < 8; ++i) V[i] = __builtin_nontemporal_load(vp + i);

  // ---- project_to_hyperboloid(x), scale folded (scale is exactly 0 or 1) ---
  // logical x = [t, scale * X_space];  X keeps RAW space comps, X[0].x := t
  const float s      = spacedot(X, X);           // sum of squares, space only
  const float scaled = (s < HEPS) ? HEPS : s;
  const float t      = sqrtf(1.0f + scaled);
  const float scale  = (s < HEPS) ? 0.0f : 1.0f;
  X[0].x = t;                                    // time slot now holds t

  // ---- project_to_tangent(x, v) -------------------------------------------
  // inner = -t*v0 + scale * <X_space, V_space>
  const float v0    = V[0].x;
  const float inner = fmaf(-t, v0, scale * spacedot(X, V));

  // v_proj = v + inner * x_logical  (space coeff inner*scale, time coeff inner)
  const float is = inner * scale;
#pragma unroll
  for (int i = 0; i < 8; ++i) V[i] = v4_fma(is, X[i], V[i]);
  const float p0 = fmaf(inner, t, v0);           // correct time component
  V[0].x = p0;

  // check = mink(x_logical, v_proj); correct if |check| > EPS
  const float check = fmaf(-t, p0, scale * spacedot(X, V));
  const float d  = (fabsf(check) > HEPS) ? check : 0.0f;
  const float ds = d * scale;
#pragma unroll
  for (int i = 0; i < 8; ++i) V[i] = v4_fma(-ds, X[i], V[i]);
  V[0].x = fmaf(-d, t, p0);

  // ---- vn = minkowski_norm(v) ---------------------------------------------
  const float ivv = mink(V, V);
  const float a   = fabsf(ivv);
  const float rt  = sqrtf(a);
  const float st  = (a < HEPS) ? 0.5f * rt : rt;
  const float vn  = (st > HMAXN) ? HMAXN : st;

  // coeffs: cosh/sinh via single exp + rcp (vn in [0,20], no overflow)
  const bool  z  = vn < HEPS;
  const float e  = __expf(vn);
  const float ei = __builtin_amdgcn_rcpf(e);
  const float c1 = z ? 1.0f : 0.5f * (e + ei);
  const float c2 = z ? 0.0f : (0.5f * (e - ei)) / fmaxf(vn, HEPS);

  // ---- result = c1*x_logical + c2*v ---------------------------------------
  const float c1s = c1 * scale;                  // space coeff (exact fold)
  const float r0  = fmaf(c2, V[0].x, c1 * t);    // time component
#pragma unroll
  for (int i = 0; i < 8; ++i) X[i] = v4_ab(c2, V[i], c1s, X[i]);
  X[0].x = r0;

  // ---- final project_to_hyperboloid (must materialize: values are stored) -
  const float s2      = spacedot(X, X);
  const float scaled2 = (s2 < HEPS) ? HEPS : s2;
  const float t2      = sqrtf(1.0f + scaled2);
  const float scale2  = (s2 < HEPS) ? 0.0f : 1.0f;
#pragma unroll
  for (int i = 0; i < 8; ++i) X[i] = v4_mul(X[i], scale2);
  X[0].x = t2;

#pragma unroll
  for (int i = 0; i < 8; ++i) __builtin_nontemporal_store(X[i], op + i);
}

extern "C" void kernel_launch(void* const* d_in, const int* in_sizes, int n_in,
                              void* d_out, int out_size, void* d_ws, size_t ws_size,
                              hipStream_t stream) {
  (void)d_ws; (void)ws_size; (void)out_size; (void)n_in;
  const float* x = (const float*)d_in[0];
  const float* v = (const float*)d_in[1];
  float* out     = (float*)d_out;

  const long long nrows = (long long)in_sizes[0] / 32;  // D = 32
  const int  block = 256;                                // 8 wave32 per block
  const long long grid = (nrows + block - 1) / block;

  hyperbolic_expmap_kernel<<<dim3((unsigned)grid), dim3(block), 0, stream>>>(
      x, v, out, nrows);
}